// KrigingLoss_77421080477995
// MI455X (gfx1250) — compile-verified
//
#include <hip/hip_runtime.h>
#include <math.h>

// Problem constants (fixed by the reference's setup_inputs)
#define BC    1024        // B*C = 16*64
#define S     256         // SAMPLE
#define W1    128         // 2*W
#define HW    4096        // H*W
#define H1W1  16384       // H1*W1
#define NLAGS 6
#define EPS   1e-7f

typedef __attribute__((ext_vector_type(2))) float v2f;
typedef __attribute__((ext_vector_type(8))) float v8f;

// ---- workspace layout (element offsets into a float*/int* view; all 4B) ----
#define WS_X0S   0u            // [BC][S]  f32  gathered x0 samples
#define WS_IU    262144u       // [BC][S]  i32  pooled row coords
#define WS_JU    524288u       // [BC][S]  i32  pooled col coords
#define WS_X     786432u       // [S][BC]  f32  x1 samples, s-major (WMMA B operand)
#define WS_XT    1048576u      // [BC][S]  f32  x1 samples, bc-major (column dots)
#define WS_M     1310720u      // [8][S][S] f32 level-1 bin masks (0/1)
#define WS_R     1835008u      // [8][S]   f32  mask row sums
#define WS_BMIN  1837056u      // [BC]     i32  per-channel min d^2
#define WS_BMAX  1838080u      // [BC]     i32  per-channel max d^2
#define WS_BINS  1839104u      // [8]      f32  7 bin edges
#define WS_CNT   1839112u      // [8]      f32  level-1 per-bin pair counts
#define WS_SV0   1839120u      // [BC][8]  f32
#define WS_SV1   1847312u      // [BC][8]  f32
#define WS_END   1855504u      // ~7.42 MB total

// ---------------------------------------------------------------------------
// 1) Gather samples + coordinates; build X (s-major) and XT (bc-major).
// ---------------------------------------------------------------------------
__global__ void k_gather(const float* __restrict__ x0, const int* __restrict__ ind,
                         const float* __restrict__ x1, const int* __restrict__ perm0,
                         const int* __restrict__ perm1, float* __restrict__ ws)
{
    int bc = blockIdx.x, s = threadIdx.x;
    int* wsi = (int*)ws;
    int p0 = perm0[s];
    ws [WS_X0S + bc * S + s] = x0[bc * HW + p0];
    int idx = ind[bc * HW + p0];
    wsi[WS_IU + bc * S + s] = idx / W1;
    wsi[WS_JU + bc * S + s] = idx % W1;
    int p1 = perm1[s];
    float v1 = x1[bc * H1W1 + p1];
    ws[WS_XT + bc * S + s] = v1;
    ws[WS_X  + s * BC + bc] = v1;
}

// ---------------------------------------------------------------------------
// 2) Per-channel min/max of integer squared distance (sqrt is monotone, and
//    all squared coord diffs are exactly representable -> bitwise-faithful).
// ---------------------------------------------------------------------------
__global__ void k_minmax(float* __restrict__ ws)
{
    __shared__ int liu[S], lju[S];
    __shared__ int rmin[S], rmax[S];
    int bc = blockIdx.x, t = threadIdx.x;
    int* wsi = (int*)ws;
    liu[t] = wsi[WS_IU + bc * S + t];
    lju[t] = wsi[WS_JU + bc * S + t];
    __syncthreads();
    int iu_s = liu[t], ju_s = lju[t];
    int mn = 0x7fffffff, mx = -1;
    for (int u = 0; u < S; ++u) {
        int di = iu_s - liu[u], dj = ju_s - lju[u];
        int d2 = di * di + dj * dj;
        mn = min(mn, d2); mx = max(mx, d2);
    }
    rmin[t] = mn; rmax[t] = mx; __syncthreads();
    for (int off = 128; off; off >>= 1) {
        if (t < off) { rmin[t] = min(rmin[t], rmin[t + off]);
                       rmax[t] = max(rmax[t], rmax[t + off]); }
        __syncthreads();
    }
    if (!t) { wsi[WS_BMIN + bc] = rmin[0]; wsi[WS_BMAX + bc] = rmax[0]; }
}

// ---------------------------------------------------------------------------
// 3) Global min/max -> bin edges; also zero the level-1 count accumulators
//    (keeps every replay deterministic without host-side memset).
// ---------------------------------------------------------------------------
__global__ void k_bins(float* __restrict__ ws)
{
    __shared__ int rmin[256], rmax[256];
    int t = threadIdx.x;
    int* wsi = (int*)ws;
    int mn = 0x7fffffff, mx = -1;
    for (int i = t; i < BC; i += 256) {
        mn = min(mn, wsi[WS_BMIN + i]);
        mx = max(mx, wsi[WS_BMAX + i]);
    }
    rmin[t] = mn; rmax[t] = mx; __syncthreads();
    for (int off = 128; off; off >>= 1) {
        if (t < off) { rmin[t] = min(rmin[t], rmin[t + off]);
                       rmax[t] = max(rmax[t], rmax[t + off]); }
        __syncthreads();
    }
    if (!t) {
        float dmin = sqrtf((float)rmin[0]);
        float dmax = sqrtf((float)rmax[0]);
        float dd = (dmax - dmin) / (float)NLAGS;
        for (int n = 0; n < NLAGS; ++n) ws[WS_BINS + n] = dmin + dd * (float)n;
        ws[WS_BINS + NLAGS] = dmax + 0.001f;
    }
    if (t < 8) ws[WS_CNT + t] = 0.f;
}

// ---------------------------------------------------------------------------
// 4) Level-0: per-channel pair binning (per-channel masks, so VALU is the
//    right pipe). Register bin accumulators, LDS atomics for the fold.
// ---------------------------------------------------------------------------
__global__ void k_level0(float* __restrict__ ws)
{
    __shared__ int   liu[S], lju[S];
    __shared__ float lx[S];
    __shared__ float accg[NLAGS];
    __shared__ int   accc[NLAGS];
    int bc = blockIdx.x, t = threadIdx.x;
    int* wsi = (int*)ws;
    liu[t] = wsi[WS_IU + bc * S + t];
    lju[t] = wsi[WS_JU + bc * S + t];
    lx [t] = ws [WS_X0S + bc * S + t];
    if (t < NLAGS) { accg[t] = 0.f; accc[t] = 0; }
    float bins[NLAGS + 1];
    #pragma unroll
    for (int n = 0; n <= NLAGS; ++n) bins[n] = ws[WS_BINS + n];
    __syncthreads();
    int iu_s = liu[t], ju_s = lju[t];
    float xs = lx[t];
    float g6[NLAGS] = {0.f, 0.f, 0.f, 0.f, 0.f, 0.f};
    int   c6[NLAGS] = {0, 0, 0, 0, 0, 0};
    for (int u = 0; u < S; ++u) {
        float di = (float)(iu_s - liu[u]), dj = (float)(ju_s - lju[u]);
        float d = sqrtf(di * di + dj * dj);
        float dx = xs - lx[u];
        float g = 0.5f * dx * dx;
        #pragma unroll
        for (int n = 0; n < NLAGS; ++n) {
            bool m = (d >= bins[n]) && (d < bins[n + 1]);
            if (m) { g6[n] += g; c6[n] += 1; }
        }
    }
    #pragma unroll
    for (int n = 0; n < NLAGS; ++n) {
        atomicAdd(&accg[n], g6[n]);
        atomicAdd(&accc[n], c6[n]);
    }
    __syncthreads();
    if (t < 8) {
        float sv = 0.f;                          // edge columns stay 0
        if (t >= 1 && t <= NLAGS) {
            int c = accc[t - 1];
            sv = (c == 0) ? 0.f : accg[t - 1] / ((float)c + EPS);
        }
        ws[WS_SV0 + bc * 8 + t] = sv;
    }
}

// ---------------------------------------------------------------------------
// 5) Level-1: build 8 shared 0/1 masks over d1 + row sums + counts.
// ---------------------------------------------------------------------------
__global__ void k_mask(const int* __restrict__ perm1, float* __restrict__ ws)
{
    __shared__ float red[S];
    int s = blockIdx.x, t = threadIdx.x;
    int ps = perm1[s], pt = perm1[t];
    float di = (float)(ps / W1 - pt / W1);
    float dj = (float)(ps % W1 - pt % W1);
    float d = sqrtf(di * di + dj * dj);
    float bins[NLAGS + 1];
    #pragma unroll
    for (int n = 0; n <= NLAGS; ++n) bins[n] = ws[WS_BINS + n];
    for (int col = 0; col < 8; ++col) {
        bool m;
        if (col == 0)      m = (d < bins[0]);
        else if (col == 7) m = (d >= bins[NLAGS]);
        else               m = (d >= bins[col - 1]) && (d < bins[col]);
        float mv = m ? 1.f : 0.f;
        ws[WS_M + (unsigned)col * S * S + s * S + t] = mv;
        red[t] = mv; __syncthreads();
        for (int off = 128; off; off >>= 1) {
            if (t < off) red[t] += red[t + off];
            __syncthreads();
        }
        if (!t) {
            ws[WS_R + col * S + s] = red[0];
            atomicAdd(&ws[WS_CNT + col], red[0]);
        }
        __syncthreads();
    }
}

// ---------------------------------------------------------------------------
// 6) Level-1 bilinear forms on the matrix pipe:
//      sv1[bc][n] = (sum_s x_s^2 R_n[s] - x^T M_n x) / (cnt_n + EPS)
//    x^T M_n x for 1024 channels == column dots against Y = M_n * X, computed
//    as 16x16 f32 tiles with V_WMMA_F32_16X16X4_F32 (K accumulated over 256).
//    A layout (ISA 7.12.2, 32-bit A 16x4): lanes 0-15 hold K{0,1}, 16-31 K{2,3}.
//    B/C/D: rows striped across lanes within a VGPR.
// ---------------------------------------------------------------------------
__global__ __launch_bounds__(256) void k_level1_wmma(float* __restrict__ ws)
{
    __shared__ float t_sh[16];
    int n   = blockIdx.y;                 // bin column 0..7
    int j   = blockIdx.x;                 // bc-tile 0..63
    int tid = threadIdx.x;
    int wave = tid >> 5, lane = tid & 31;
    int lhalf = lane >> 4;                // lane group (0: lanes 0-15, 1: 16-31)
    int l15   = lane & 15;
    int bc0   = j * 16;

    const float* M  = ws + WS_M + (unsigned)n * S * S;
    const float* X  = ws + WS_X;
    const float* XT = ws + WS_XT;

    if (tid < 16) t_sh[tid] = 0.f;
    __syncthreads();

    int col = bc0 + l15;
    float p = 0.f;                        // per-lane partial of x^T M x
    for (int mi = 0; mi < 2; ++mi) {      // 8 waves x 2 = 16 m-tiles
        int s0 = (wave * 2 + mi) * 16;
        v8f c = {0.f, 0.f, 0.f, 0.f, 0.f, 0.f, 0.f, 0.f};
        const float* Ar = M + (s0 + l15) * S + lhalf * 2;   // A: row s0+l15, K base
        const float* Bc = X + col + (unsigned)(lhalf * 2) * BC; // B: K rows striped
        for (int k = 0; k < S; k += 4) {
            v2f a; a.x = Ar[k];                 a.y = Ar[k + 1];
            v2f b; b.x = Bc[(unsigned)k * BC];  b.y = Bc[(unsigned)(k + 1) * BC];
            c = __builtin_amdgcn_wmma_f32_16x16x4_f32(
                    false, a, false, b, (short)0, c, false, false);
        }
        // fold this D tile into the column dot: lanes 0-15 hold rows s0..s0+7,
        // lanes 16-31 rows s0+8..s0+15 (C/D layout), same column per lane pair.
        const float* xc = XT + col * S + s0 + lhalf * 8;
        float q = 0.f;
        #pragma unroll
        for (int r = 0; r < 8; ++r) q += xc[r] * c[r];
        p += q;
    }
    p += __shfl_xor(p, 16, 32);           // combine the two row-halves
    if (lane < 16) atomicAdd(&t_sh[l15], p);
    __syncthreads();

    if (tid < 16) {
        int colw = bc0 + tid;
        const float* xc = XT + colw * S;
        const float* Rr = ws + WS_R + n * S;
        float q = 0.f;
        for (int s = 0; s < S; ++s) { float v = xc[s]; q += v * v * Rr[s]; }
        float cnt = ws[WS_CNT + n];
        float sum_g = q - t_sh[tid];
        float sv = (cnt == 0.f) ? 0.f : sum_g / (cnt + EPS);
        ws[WS_SV1 + colw * 8 + n] = sv;
    }
}

// ---------------------------------------------------------------------------
// 7) Final loss: mean over BC*8 of (sv1 - sv0)^2.
// ---------------------------------------------------------------------------
__global__ void k_loss(const float* __restrict__ ws, float* __restrict__ out)
{
    __shared__ float red[256];
    int t = threadIdx.x;
    float acc = 0.f;
    for (int i = t; i < BC * 8; i += 256) {
        float dsv = ws[WS_SV1 + i] - ws[WS_SV0 + i];
        acc += dsv * dsv;
    }
    red[t] = acc; __syncthreads();
    for (int off = 128; off; off >>= 1) {
        if (t < off) red[t] += red[t + off];
        __syncthreads();
    }
    if (!t) out[0] = red[0] / (float)(BC * 8);
}

// ---------------------------------------------------------------------------
extern "C" void kernel_launch(void* const* d_in, const int* in_sizes, int n_in,
                              void* d_out, int out_size, void* d_ws, size_t ws_size,
                              hipStream_t stream)
{
    const float* x0    = (const float*)d_in[0];
    const int*   ind   = (const int*)  d_in[1];
    const float* x1    = (const float*)d_in[2];
    const int*   perm0 = (const int*)  d_in[3];
    const int*   perm1 = (const int*)  d_in[4];
    float* ws  = (float*)d_ws;
    float* out = (float*)d_out;
    (void)in_sizes; (void)n_in; (void)out_size; (void)ws_size;

    k_gather<<<BC, S, 0, stream>>>(x0, ind, x1, perm0, perm1, ws);
    k_minmax<<<BC, S, 0, stream>>>(ws);
    k_bins<<<1, 256, 0, stream>>>(ws);
    k_level0<<<BC, S, 0, stream>>>(ws);
    k_mask<<<S, S, 0, stream>>>(perm1, ws);
    dim3 g1(64, 8);
    k_level1_wmma<<<g1, 256, 0, stream>>>(ws);
    k_loss<<<1, 256, 0, stream>>>(ws, out);
}